// GraphTransGeoGCN_78546361909449
// MI455X (gfx1250) — compile-verified
//
#include <hip/hip_runtime.h>
#include <hip/hip_bf16.h>

typedef __attribute__((ext_vector_type(2))) float v2f;
typedef __attribute__((ext_vector_type(8))) float v8f;

#define N_NODES  100000
#define N_EDGES  1280000
#define DIM      64
#define N_GRAPHS 1000
#define BN_EPS   1e-5f

// ---------------- utility ----------------
__global__ void k_fill_f32(float* __restrict__ p, float v, int n) {
    int i = blockIdx.x * blockDim.x + threadIdx.x;
    if (i < n) p[i] = v;
}

// degree over targets (self-loop handled by deg init = 1.0)
__global__ void k_deg_edges(const long long* __restrict__ ei, float* __restrict__ deg, int ne) {
    int e = blockIdx.x * blockDim.x + threadIdx.x;
    if (e < ne) atomicAdd(&deg[ei[(size_t)ne + e]], 1.0f);
}

__global__ void k_rsqrt(const float* __restrict__ deg, float* __restrict__ dinv, int n) {
    int i = blockIdx.x * blockDim.x + threadIdx.x;
    if (i < n) {
        float d = deg[i];
        dinv[i] = (d > 0.0f) ? rsqrtf(d) : 0.0f;
    }
}

// per-column sum / sumsq (optionally relu-in-place first). blockDim = 64 (tid = column).
__global__ void k_colstats(float* __restrict__ data, float* __restrict__ stats,
                           int n, int apply_relu) {
    int c = threadIdx.x;
    int start = blockIdx.x * 256;
    int end = start + 256; if (end > n) end = n;
    float s = 0.0f, ss = 0.0f;
    for (int i = start; i < end; ++i) {
        float v = data[(size_t)i * DIM + c];
        if (apply_relu) {
            v = fmaxf(v, 0.0f);
            data[(size_t)i * DIM + c] = v;
        }
        s += v; ss += v * v;
    }
    atomicAdd(&stats[c], s);
    atomicAdd(&stats[DIM + c], ss);
}

// fold BN (from stats) into conv weight: Wmod[k][j] = s_k * W[k][j], cvec[j] = sum_k t_k W[k][j]
__global__ void k_fold_bn(const float* __restrict__ stats,
                          const float* __restrict__ gamma, const float* __restrict__ beta,
                          const float* __restrict__ W,
                          float* __restrict__ Wmod, float* __restrict__ cvec, float invN) {
    __shared__ float sS[DIM], sT[DIM];
    int t = threadIdx.x; // 256 threads
    if (t < DIM) {
        float m   = stats[t] * invN;
        float var = stats[DIM + t] * invN - m * m;
        float sc  = rsqrtf(var + BN_EPS) * gamma[t];
        sS[t] = sc;
        sT[t] = beta[t] - m * sc;
    }
    __syncthreads();
    for (int i = t; i < DIM * DIM; i += 256) {
        int k = i >> 6;
        Wmod[i] = sS[k] * W[i];
    }
    if (t < DIM) {
        float acc = 0.0f;
        for (int k = 0; k < DIM; ++k) acc += sT[k] * W[k * DIM + t];
        cvec[t] = acc;
    }
}

// ---------------- FP32 WMMA GEMM: out[N,64] = X[N,64] @ Wm[64,64] + cvec ----------------
// blockDim=256 (8 waves). Each wave: one 16x16 tile. Block covers 32 rows x 64 cols.
// N_NODES = 100000 = 3125 * 32, exact -> EXEC all-ones for every WMMA.
__global__ void k_gemm64_wmma(const float* __restrict__ X, const float* __restrict__ Wm,
                              const float* __restrict__ cvec, float* __restrict__ out) {
    __shared__ float Wl[DIM * DIM];
    __shared__ float cv[DIM];
    int t = threadIdx.x;
    for (int i = t; i < DIM * DIM; i += 256) Wl[i] = Wm[i];
    if (t < DIM) cv[t] = cvec[t];
    __syncthreads();

    int wave = t >> 5, lane = t & 31;
    int row0 = blockIdx.x * 32 + (wave >> 2) * 16;  // 2 row-tiles
    int col0 = (wave & 3) * 16;                     // 4 col-tiles
    int mlane = lane & 15;
    int khalf = (lane >> 4) * 2;                    // lanes 0-15: K=0,1 ; lanes 16-31: K=2,3

    const float* xrow = X + (size_t)(row0 + mlane) * DIM;
    v8f acc = {};
    #pragma unroll
    for (int kk = 0; kk < DIM; kk += 4) {
        v2f a, b;
        a.x = xrow[kk + khalf];
        a.y = xrow[kk + khalf + 1];
        b.x = Wl[(kk + khalf) * DIM + col0 + mlane];
        b.y = Wl[(kk + khalf + 1) * DIM + col0 + mlane];
        acc = __builtin_amdgcn_wmma_f32_16x16x4_f32(false, a, false, b,
                                                    (short)0, acc, false, false);
    }
    // D layout: VGPR r -> lanes 0-15: row row0+r ; lanes 16-31: row row0+8+r ; col = col0 + (lane&15)
    int srow = row0 + ((lane >> 4) ? 8 : 0);
    int scol = col0 + mlane;
    float add = cv[scol];
    float* o = out + (size_t)srow * DIM + scol;
    #pragma unroll
    for (int r = 0; r < 8; ++r) o[(size_t)r * DIM] = acc[r] + add;
}

// out[i] = dinv[i]^2 * H[i] + bias  (self-loop + conv bias; also initializes agg buffer)
__global__ void k_init_self(const float* __restrict__ H, const float* __restrict__ dinv,
                            const float* __restrict__ bias, float* __restrict__ out, int n) {
    int gid = blockIdx.x * blockDim.x + threadIdx.x;
    int i = gid >> 4;
    if (i >= n) return;
    int q = (gid & 15) * 4;
    float d = dinv[i]; d = d * d;
    float4 h = *(const float4*)(H + (size_t)i * DIM + q);
    float4 o;
    o.x = d * h.x + bias[q + 0];
    o.y = d * h.y + bias[q + 1];
    o.z = d * h.z + bias[q + 2];
    o.w = d * h.w + bias[q + 3];
    *(float4*)(out + (size_t)i * DIM + q) = o;
}

// per-edge scatter: out[col] += dinv[row]*dinv[col] * H[row].  16 threads / edge, float4 each.
__global__ void k_edge_scatter(const long long* __restrict__ ei, const float* __restrict__ dinv,
                               const float* __restrict__ H, float* __restrict__ out, int ne) {
    int gid = blockIdx.x * blockDim.x + threadIdx.x;
    int e = gid >> 4;
    if (e >= ne) return;
    int q = (gid & 15) * 4;
    long long r = ei[e];
    long long c = ei[(size_t)ne + e];
    float nm = dinv[r] * dinv[c];
    float4 h = *(const float4*)(H + (size_t)r * DIM + q);
    float* o = out + (size_t)c * DIM + q;
    atomicAdd(o + 0, nm * h.x);
    atomicAdd(o + 1, nm * h.y);
    atomicAdd(o + 2, nm * h.z);
    atomicAdd(o + 3, nm * h.w);
}

// per-graph pooling of raw relu values (final BN affine applied in epilogue). blockDim=64.
__global__ void k_pool(const float* __restrict__ R, const long long* __restrict__ batch,
                       float* __restrict__ psum, float* __restrict__ pcnt, int n) {
    int c = threadIdx.x;
    int start = blockIdx.x * 256;
    int end = start + 256; if (end > n) end = n;
    for (int i = start; i < end; ++i) {
        long long g = batch[i];
        atomicAdd(&psum[(size_t)g * DIM + c], R[(size_t)i * DIM + c]);
        if (c == 0) atomicAdd(&pcnt[g], 1.0f);
    }
}

// out[g] = ((psum[g]/cnt) * s + t) @ Wout + bout   (BN affine commutes with the mean)
__global__ void k_final(const float* __restrict__ psum, const float* __restrict__ pcnt,
                        const float* __restrict__ stats,
                        const float* __restrict__ gamma, const float* __restrict__ beta,
                        const float* __restrict__ Wout, const float* __restrict__ bout,
                        float* __restrict__ out, float invN) {
    int g = blockIdx.x * blockDim.x + threadIdx.x;
    if (g >= N_GRAPHS) return;
    float cnt = fmaxf(pcnt[g], 1.0f);
    float o0 = bout[0], o1 = bout[1];
    for (int c = 0; c < DIM; ++c) {
        float m   = stats[c] * invN;
        float var = stats[DIM + c] * invN - m * m;
        float sc  = rsqrtf(var + BN_EPS) * gamma[c];
        float tt  = beta[c] - m * sc;
        float v   = (psum[(size_t)g * DIM + c] / cnt) * sc + tt;
        o0 += v * Wout[c * 2 + 0];
        o1 += v * Wout[c * 2 + 1];
    }
    out[g * 2 + 0] = o0;
    out[g * 2 + 1] = o1;
}

// ---------------- launch ----------------
extern "C" void kernel_launch(void* const* d_in, const int* in_sizes, int n_in,
                              void* d_out, int out_size, void* d_ws, size_t ws_size,
                              hipStream_t stream) {
    const float*     x     = (const float*)d_in[0];
    const long long* ei    = (const long long*)d_in[1];
    const long long* batch = (const long long*)d_in[2];
    const float* bn_in_g = (const float*)d_in[3];
    const float* bn_in_b = (const float*)d_in[4];
    const float* W1  = (const float*)d_in[5];
    const float* b1  = (const float*)d_in[6];
    const float* g1  = (const float*)d_in[7];
    const float* be1 = (const float*)d_in[8];
    const float* W2  = (const float*)d_in[9];
    const float* b2  = (const float*)d_in[10];
    const float* g2  = (const float*)d_in[11];
    const float* be2 = (const float*)d_in[12];
    const float* Wout = (const float*)d_in[13];
    const float* bout = (const float*)d_in[14];
    float* out = (float*)d_out;

    // workspace layout (floats)
    const size_t NB = (size_t)N_NODES * DIM;     // 6.4M floats per big buffer
    float* A     = (float*)d_ws;                 // GEMM output / message features
    float* B     = A + NB;                       // aggregation output / relu features
    float* deg   = B + NB;                       // N_NODES
    float* dinv  = deg + N_NODES;                // N_NODES
    float* stats = dinv + N_NODES;               // 128 (sum, sumsq)
    float* Wmod  = stats + 128;                  // 64*64
    float* cvec  = Wmod + DIM * DIM;             // 64
    float* psum  = cvec + DIM;                   // N_GRAPHS*64
    float* pcnt  = psum + (size_t)N_GRAPHS * DIM;// N_GRAPHS

    const float invN = 1.0f / (float)N_NODES;
    const int B256 = 256;
    const int gNodes   = (N_NODES + 255) / 256;
    const int gEdges   = (N_EDGES + 255) / 256;
    const int gStat    = (N_NODES + 255) / 256;        // blockDim 64, 256 nodes/block
    const int gNodeF4  = (N_NODES * 16 + 255) / 256;
    const int gEdgeF4  = (N_EDGES * 16 + 255) / 256;

    // --- degree / dinv (self-loops via init=1.0) ---
    k_fill_f32<<<gNodes, B256, 0, stream>>>(deg, 1.0f, N_NODES);
    k_deg_edges<<<gEdges, B256, 0, stream>>>(ei, deg, N_EDGES);
    k_rsqrt<<<gNodes, B256, 0, stream>>>(deg, dinv, N_NODES);

    // --- layer 1: fold input-BN into W1 ---
    k_fill_f32<<<1, 128, 0, stream>>>(stats, 0.0f, 128);
    k_colstats<<<gStat, 64, 0, stream>>>((float*)x, stats, N_NODES, 0);
    k_fold_bn<<<1, 256, 0, stream>>>(stats, bn_in_g, bn_in_b, W1, Wmod, cvec, invN);
    k_gemm64_wmma<<<N_NODES / 32, 256, 0, stream>>>(x, Wmod, cvec, A);
    k_init_self<<<gNodeF4, B256, 0, stream>>>(A, dinv, b1, B, N_NODES);
    k_edge_scatter<<<gEdgeF4, B256, 0, stream>>>(ei, dinv, A, B, N_EDGES);

    // --- layer 2: relu + BN stats, fold into W2 ---
    k_fill_f32<<<1, 128, 0, stream>>>(stats, 0.0f, 128);
    k_colstats<<<gStat, 64, 0, stream>>>(B, stats, N_NODES, 1);   // B now holds relu(h1)
    k_fold_bn<<<1, 256, 0, stream>>>(stats, g1, be1, W2, Wmod, cvec, invN);
    k_gemm64_wmma<<<N_NODES / 32, 256, 0, stream>>>(B, Wmod, cvec, A);
    k_init_self<<<gNodeF4, B256, 0, stream>>>(A, dinv, b2, B, N_NODES);
    k_edge_scatter<<<gEdgeF4, B256, 0, stream>>>(ei, dinv, A, B, N_EDGES);

    // --- final: relu + BN stats, per-graph mean pool, BN affine + Wout ---
    k_fill_f32<<<1, 128, 0, stream>>>(stats, 0.0f, 128);
    k_fill_f32<<<(N_GRAPHS * DIM + 255) / 256, B256, 0, stream>>>(psum, 0.0f, N_GRAPHS * DIM);
    k_fill_f32<<<(N_GRAPHS + 255) / 256, B256, 0, stream>>>(pcnt, 0.0f, N_GRAPHS);
    k_colstats<<<gStat, 64, 0, stream>>>(B, stats, N_NODES, 1);   // B now holds relu(h2)
    k_pool<<<gStat, 64, 0, stream>>>(B, batch, psum, pcnt, N_NODES);
    k_final<<<(N_GRAPHS + 127) / 128, 128, 0, stream>>>(psum, pcnt, stats, g2, be2,
                                                        Wout, bout, out, invN);
    (void)in_sizes; (void)n_in; (void)out_size; (void)ws_size;
}